// MANN_35751307772252
// MI455X (gfx1250) — compile-verified
//
#include <hip/hip_runtime.h>

#define BS 64
#define SEQ 128
#define IN_DIM 256
#define HDIM 512
#define NSLOT 4096
#define MDIM 128
#define RHEAD 4
#define OUTD 64
#define GAMMA 0.95f
#define KLSTM (IN_DIM + 1 + HDIM)   /* 769 */
#define KPAD 772                    /* KLSTM padded to multiple of 4 */
#define G4H (4 * HDIM)              /* 2048 */
#define OUTW (HDIM + RHEAD * MDIM)  /* 1024 */

typedef __attribute__((ext_vector_type(2))) float v2f;
typedef __attribute__((ext_vector_type(8))) float v8f;

__device__ __forceinline__ v8f wmma_k4(v2f a, v2f b, v8f c) {
  // V_WMMA_F32_16X16X4_F32 : D = A(16x4,f32) * B(4x16,f32) + C(16x16,f32)
  return __builtin_amdgcn_wmma_f32_16x16x4_f32(
      /*neg_a=*/false, a, /*neg_b=*/false, b,
      /*c_mod=*/(short)0, c, /*reuse_a=*/false, /*reuse_b=*/false);
}

__device__ __forceinline__ float sigmoidf(float x) {
  return 1.0f / (1.0f + __expf(-x));
}

// ---- one-time: zero-pad Wl[769,2048] -> Wlp[772,2048] ----
__global__ __launch_bounds__(256)
void k_pad_wl(const float* __restrict__ Wl, float* __restrict__ Wlp) {
  const size_t i = (size_t)blockIdx.x * 256 + threadIdx.x;  // over KPAD*G4H
  const int k = (int)(i / G4H), c = (int)(i % G4H);
  Wlp[i] = (k < KLSTM) ? Wl[(size_t)k * G4H + c] : 0.0f;
}

// ---- per step: build A[64,772] = [x_t | y_prev | hidden | 0pad] (dense, branch-free GEMM after) ----
__global__ __launch_bounds__(256)
void k_build_a(const float* __restrict__ x, const float* __restrict__ y,
               const float* __restrict__ hidden, float* __restrict__ A, int t) {
  const int b = blockIdx.x;
  float* arow = A + (size_t)b * KPAD;
  for (int k = threadIdx.x; k < KPAD; k += 256) {
    float v;
    if (k < IN_DIM)       v = x[((size_t)b * SEQ + t) * IN_DIM + k];
    else if (k == IN_DIM) v = (t == 0) ? 0.0f : y[(size_t)b * SEQ + (t - 1)];
    else {
      const int h = k - (IN_DIM + 1);
      v = (h < HDIM) ? hidden[b * HDIM + h] : 0.0f;
    }
    arow[k] = v;
  }
}

// gates[64,2048] = A[64,772] @ Wlp[772,2048] + bl  (fp32 WMMA, branch-free inner loop)
__global__ __launch_bounds__(32)
void k_lstm_gates(const float* __restrict__ A, const float* __restrict__ Wlp,
                  const float* __restrict__ bl, float* __restrict__ gates) {
  const int lane = threadIdx.x;             // wave32
  const int rt = blockIdx.x;                // batch-row tile (0..3)
  const int ct = blockIdx.y;                // col tile (0..127)
  const int row  = lane & 15;
  const int koff = (lane >> 4) << 1;        // lanes 0-15 -> k+0/1, 16-31 -> k+2/3
  const int colg = ct * 16 + (lane & 15);
  const float* ar = A + (size_t)(rt * 16 + row) * KPAD;
  v8f c = {};
  for (int k0 = 0; k0 < KPAD; k0 += 4) {
    const int ka = k0 + koff;
    const v2f a = *(const v2f*)(ar + ka);          // contiguous K-pair -> b64
    v2f w;
    w.x = Wlp[(size_t)ka * G4H + colg];            // column-strided (coalesced across lanes)
    w.y = Wlp[(size_t)(ka + 1) * G4H + colg];
    c = wmma_k4(a, w, c);
  }
  const float bias = bl[colg];
  const int rbase = rt * 16 + ((lane < 16) ? 0 : 8);
#pragma unroll
  for (int j = 0; j < 8; ++j)
    gates[(size_t)(rbase + j) * G4H + colg] = c[j] + bias;
}

// LSTM pointwise: f,i,o,u -> cell,hidden; hidden also into outputs[b,t,0:512]
__global__ __launch_bounds__(HDIM)
void k_lstm_point(const float* __restrict__ gates, float* __restrict__ hidden,
                  float* __restrict__ cell, float* __restrict__ outputs, int t) {
  const int b = blockIdx.x, h = threadIdx.x;
  const float* g = gates + (size_t)b * G4H;
  const float f = sigmoidf(g[h]);
  const float i = sigmoidf(g[HDIM + h]);
  const float o = sigmoidf(g[2 * HDIM + h]);
  const float u = tanhf(g[3 * HDIM + h]);
  const float cn = f * cell[b * HDIM + h] + i * u;
  cell[b * HDIM + h] = cn;
  const float hn = o * tanhf(cn);
  hidden[b * HDIM + h] = hn;
  outputs[((size_t)b * SEQ + t) * OUTW + h] = hn;
}

// Per-batch: find R least-used slots (stable ascending); zero the norm accumulators.
__global__ __launch_bounds__(BS)
void k_prep(const float* __restrict__ uw, int* __restrict__ idx,
            float* __restrict__ norms) {
  const int b = threadIdx.x;
  if (b == 0) { norms[0] = 0.0f; norms[1] = 0.0f; }
  float bv[RHEAD]; int bi[RHEAD];
#pragma unroll
  for (int r = 0; r < RHEAD; ++r) { bv[r] = 3.4e38f; bi[r] = 0; }
  const float* u = uw + (size_t)b * NSLOT;
  for (int n = 0; n < NSLOT; ++n) {
    const float v = u[n];
    if (v < bv[RHEAD - 1]) {
      int p = RHEAD - 1;
      while (p > 0 && v < bv[p - 1]) { bv[p] = bv[p - 1]; bi[p] = bi[p - 1]; --p; }
      bv[p] = v; bi[p] = n;
    }
  }
#pragma unroll
  for (int r = 0; r < RHEAD; ++r) idx[b * RHEAD + r] = bi[r];
}

// query[64,512] = hidden @ Wm + bm (fp32 WMMA); fused ||query||^2 accumulation.
__global__ __launch_bounds__(32)
void k_query(const float* __restrict__ hidden, const float* __restrict__ Wm,
             const float* __restrict__ bm, float* __restrict__ query,
             float* __restrict__ norms) {
  const int lane = threadIdx.x;
  const int rt = blockIdx.x;                 // 0..3
  const int ct = blockIdx.y;                 // 0..31
  const int row  = lane & 15;
  const int koff = (lane >> 4) << 1;
  const int colg = ct * 16 + (lane & 15);
  const float* ar = hidden + (size_t)(rt * 16 + row) * HDIM;
  v8f c = {};
  for (int k0 = 0; k0 < HDIM; k0 += 4) {
    const int ka = k0 + koff;
    const v2f a = *(const v2f*)(ar + ka);
    v2f w;
    w.x = Wm[(size_t)ka * HDIM + colg];
    w.y = Wm[(size_t)(ka + 1) * HDIM + colg];
    c = wmma_k4(a, w, c);
  }
  const float bias = bm[colg];
  const int rbase = rt * 16 + ((lane < 16) ? 0 : 8);
  float ss = 0.0f;
#pragma unroll
  for (int j = 0; j < 8; ++j) {
    const float v = c[j] + bias;
    query[(size_t)(rbase + j) * HDIM + colg] = v;
    ss += v * v;
  }
  atomicAdd(&norms[0], ss);
}

// ww = sigmoid(alpha)*rw + (1-sigmoid(alpha)) at the least-used slot
__global__ __launch_bounds__(256)
void k_wwfill(const float* __restrict__ rw, const float* __restrict__ alpha,
              const int* __restrict__ idx, float* __restrict__ ww) {
  const size_t tid = (size_t)blockIdx.x * 256 + threadIdx.x;  // over b*R*N
  const int n  = (int)(tid & (NSLOT - 1));
  const int br = (int)(tid >> 12);
  const float a = sigmoidf(alpha[br]);
  float w = a * rw[tid];
  if (n == idx[br]) w += (1.0f - a);
  ww[tid] = w;
}

// memory[b,n,:] += sum_r ww[b,r,n]*query[b,r,:]; fused ||memory||^2 accumulation.
__global__ __launch_bounds__(MDIM)
void k_memupd(float* __restrict__ mem, const float* __restrict__ ww,
              const float* __restrict__ query, float* __restrict__ norms) {
  const int bx = blockIdx.x;                 // b*NSLOT + n
  const int b = bx >> 12, n = bx & (NSLOT - 1);
  const int m = threadIdx.x;
  const float w0 = ww[((size_t)b * RHEAD + 0) * NSLOT + n];
  const float w1 = ww[((size_t)b * RHEAD + 1) * NSLOT + n];
  const float w2 = ww[((size_t)b * RHEAD + 2) * NSLOT + n];
  const float w3 = ww[((size_t)b * RHEAD + 3) * NSLOT + n];
  const float* q = query + (size_t)b * (RHEAD * MDIM);
  const size_t off = ((size_t)b * NSLOT + n) * MDIM + m;
  const float v = mem[off] + w0 * q[m] + w1 * q[MDIM + m] +
                  w2 * q[2 * MDIM + m] + w3 * q[3 * MDIM + m];
  mem[off] = v;
  __shared__ float red[MDIM];
  red[m] = v * v;
  __syncthreads();
  for (int s = MDIM / 2; s > 0; s >>= 1) {
    if (m < s) red[m] += red[m + s];
    __syncthreads();
  }
  if (m == 0) atomicAdd(&norms[1], red[0]);
}

// logits[b,r,n] = <query[b,r,:], memory[b,n,:]> / (||q||*||M||)  via fp32 WMMA.
// A tile (16x128: rows 0..3 = read heads, 4..15 = zero) staged in LDS via
// gfx1250 async global->LDS copy; single wave so s_wait_asynccnt suffices.
__global__ __launch_bounds__(32)
void k_logits(const float* __restrict__ query, const float* __restrict__ mem,
              const float* __restrict__ norms, float* __restrict__ logits) {
  __shared__ float qs[16 * MDIM];            // 8 KB padded A tile
  const int lane = threadIdx.x;
  const int b  = blockIdx.x;
  const int nt = blockIdx.y;                 // slot tile (0..255)
  // zero padding rows 4..15
  for (int i = RHEAD * MDIM + lane; i < 16 * MDIM; i += 32) qs[i] = 0.0f;
  // async-stage 4 real query rows (512 floats = 2 KB) as 128 x b128 transfers
  {
    const float* src = query + (size_t)b * (RHEAD * MDIM);
    const uint32_t lds_base = (uint32_t)(uintptr_t)(&qs[0]);
    const uint64_t gbase = (uint64_t)(uintptr_t)src;
#pragma unroll
    for (int i = 0; i < 4; ++i) {
      const uint32_t off = (uint32_t)(lane + i * 32) * 16u;
      const uint32_t ldsa = lds_base + off;
      const uint64_t ga = gbase + off;
      asm volatile("global_load_async_to_lds_b128 %0, %1, off"
                   :: "v"(ldsa), "v"(ga) : "memory");
    }
    asm volatile("s_wait_asynccnt 0x0" ::: "memory");
  }
  const int row  = lane & 15;
  const int koff = (lane >> 4) << 1;
  const int ncol = nt * 16 + (lane & 15);
  const float* mb = mem + ((size_t)b * NSLOT + ncol) * MDIM;
  const float* arow = qs + row * MDIM;
  v8f c = {};
  for (int k0 = 0; k0 < MDIM; k0 += 4) {
    const int ka = k0 + koff;
    const v2f a = *(const v2f*)(arow + ka);        // ds_load_b64
    const v2f w = *(const v2f*)(mb + ka);          // global b64 (contiguous in m)
    c = wmma_k4(a, w, c);
  }
  const float inv = rsqrtf(norms[0] * norms[1]);   // 1/(||q||_F * ||M||_F)
  if (lane < 16) {
#pragma unroll
    for (int j = 0; j < RHEAD; ++j)
      logits[((size_t)b * RHEAD + j) * NSLOT + nt * 16 + lane] = c[j] * inv;
  }
}

// rw[b,r,:] = softmax_n(logits[b,r,:])
__global__ __launch_bounds__(256)
void k_softmax_rw(const float* __restrict__ logits, float* __restrict__ rw) {
  const int br = blockIdx.x, tid = threadIdx.x;
  const float* lg = logits + (size_t)br * NSLOT;
  float* out = rw + (size_t)br * NSLOT;
  __shared__ float red[256];
  float mx = -3.4e38f;
  for (int n = tid; n < NSLOT; n += 256) mx = fmaxf(mx, lg[n]);
  red[tid] = mx; __syncthreads();
  for (int s = 128; s > 0; s >>= 1) {
    if (tid < s) red[tid] = fmaxf(red[tid], red[tid + s]);
    __syncthreads();
  }
  mx = red[0]; __syncthreads();
  float sum = 0.0f;
  for (int n = tid; n < NSLOT; n += 256) sum += __expf(lg[n] - mx);
  red[tid] = sum; __syncthreads();
  for (int s = 128; s > 0; s >>= 1) {
    if (tid < s) red[tid] += red[tid + s];
    __syncthreads();
  }
  const float inv = 1.0f / red[0];
  for (int n = tid; n < NSLOT; n += 256) out[n] = __expf(lg[n] - mx) * inv;
}

// uw = GAMMA*uw + sum_r rw_new + sum_r ww
__global__ __launch_bounds__(256)
void k_uw(float* __restrict__ uw, const float* __restrict__ rw,
          const float* __restrict__ ww) {
  const size_t t = (size_t)blockIdx.x * 256 + threadIdx.x;  // b*N + n
  const int b = (int)(t >> 12), n = (int)(t & (NSLOT - 1));
  float s = 0.0f;
#pragma unroll
  for (int r = 0; r < RHEAD; ++r) {
    const size_t o = ((size_t)b * RHEAD + r) * NSLOT + n;
    s += rw[o] + ww[o];
  }
  uw[t] = GAMMA * uw[t] + s;
}

// read_vec[b,r,:] = sum_n rw[b,r,n]*memory[b,n,:] — single pass over memory per
// batch for all four heads; prefetch-ahead (global_prefetch_b8) on streamed rows.
__global__ __launch_bounds__(MDIM)
void k_read(const float* __restrict__ mem, const float* __restrict__ rw,
            float* __restrict__ outputs, int t) {
  const int b = blockIdx.x, m = threadIdx.x;
  const float* mb = mem + (size_t)b * NSLOT * MDIM;
  const float* p  = rw + (size_t)b * RHEAD * NSLOT;
  float a0 = 0.f, a1 = 0.f, a2 = 0.f, a3 = 0.f;
  for (int n = 0; n < NSLOT; ++n) {
    if (n + 8 < NSLOT)
      __builtin_prefetch(mb + (size_t)(n + 8) * MDIM + m, 0, 1);
    const float mv = mb[(size_t)n * MDIM + m];
    a0 += p[n] * mv;
    a1 += p[NSLOT + n] * mv;
    a2 += p[2 * NSLOT + n] * mv;
    a3 += p[3 * NSLOT + n] * mv;
  }
  float* o = outputs + ((size_t)b * SEQ + t) * OUTW + HDIM + m;
  o[0] = a0; o[MDIM] = a1; o[2 * MDIM] = a2; o[3 * MDIM] = a3;
}

// fin[8192,64] = outputs[8192,1024] @ Wf[1024,64] + bf  (fp32 WMMA)
__global__ __launch_bounds__(32)
void k_final_gemm(const float* __restrict__ outputs, const float* __restrict__ Wf,
                  const float* __restrict__ bfv, float* __restrict__ fin) {
  const int lane = threadIdx.x;
  const int rt = blockIdx.x;                 // 0..511
  const int ct = blockIdx.y;                 // 0..3
  const int row  = lane & 15;
  const int koff = (lane >> 4) << 1;
  const int colg = ct * 16 + (lane & 15);
  const float* ar = outputs + (size_t)(rt * 16 + row) * OUTW;
  v8f c = {};
  for (int k0 = 0; k0 < OUTW; k0 += 4) {
    const int ka = k0 + koff;
    const v2f a = *(const v2f*)(ar + ka);
    v2f w;
    w.x = Wf[(size_t)ka * OUTD + colg];
    w.y = Wf[(size_t)(ka + 1) * OUTD + colg];
    c = wmma_k4(a, w, c);
  }
  const float bias = bfv[colg];
  const int rbase = rt * 16 + ((lane < 16) ? 0 : 8);
#pragma unroll
  for (int j = 0; j < 8; ++j)
    fin[(size_t)(rbase + j) * OUTD + colg] = c[j] + bias;
}

__global__ __launch_bounds__(OUTD)
void k_final_softmax(const float* __restrict__ fin, float* __restrict__ probs) {
  const int row = blockIdx.x, t = threadIdx.x;
  __shared__ float red[OUTD];
  const float v = fin[(size_t)row * OUTD + t];
  red[t] = v; __syncthreads();
  for (int s = OUTD / 2; s > 0; s >>= 1) {
    if (t < s) red[t] = fmaxf(red[t], red[t + s]);
    __syncthreads();
  }
  const float mx = red[0]; __syncthreads();
  const float e = __expf(v - mx);
  red[t] = e; __syncthreads();
  for (int s = OUTD / 2; s > 0; s >>= 1) {
    if (t < s) red[t] += red[t + s];
    __syncthreads();
  }
  probs[(size_t)row * OUTD + t] = e / red[0];
}

extern "C" void kernel_launch(void* const* d_in, const int* in_sizes, int n_in,
                              void* d_out, int out_size, void* d_ws, size_t ws_size,
                              hipStream_t stream) {
  (void)in_sizes; (void)n_in; (void)out_size; (void)ws_size;
  const float* x      = (const float*)d_in[0];
  const float* y      = (const float*)d_in[1];
  const float* mem_in = (const float*)d_in[2];
  const float* rw_in  = (const float*)d_in[3];
  const float* uw_in  = (const float*)d_in[4];
  const float* Wl     = (const float*)d_in[5];
  const float* bl     = (const float*)d_in[6];
  const float* Wm     = (const float*)d_in[7];
  const float* bm     = (const float*)d_in[8];
  const float* Wf     = (const float*)d_in[9];
  const float* bfv    = (const float*)d_in[10];
  const float* alpha  = (const float*)d_in[11];

  // d_out = [probs | memory | rw | uw]; memory/rw/uw updated in place in d_out.
  float* out_probs = (float*)d_out;
  float* out_mem = out_probs + (size_t)BS * SEQ * OUTD;
  float* out_rw  = out_mem + (size_t)BS * NSLOT * MDIM;
  float* out_uw  = out_rw + (size_t)BS * RHEAD * NSLOT;

  float* w = (float*)d_ws;
  float* hidden  = w; w += (size_t)BS * HDIM;
  float* cell    = w; w += (size_t)BS * HDIM;
  float* gates   = w; w += (size_t)BS * G4H;
  float* query   = w; w += (size_t)BS * HDIM;
  float* Abuf    = w; w += (size_t)BS * KPAD;
  float* Wlp     = w; w += (size_t)KPAD * G4H;
  float* wwb     = w; w += (size_t)BS * RHEAD * NSLOT;
  float* logits  = w; w += (size_t)BS * RHEAD * NSLOT;
  float* outputs = w; w += (size_t)BS * SEQ * OUTW;
  float* fin     = w; w += (size_t)BS * SEQ * OUTD;
  float* norms   = w; w += 2;
  int*   idx     = (int*)w;

  hipMemcpyAsync(out_mem, mem_in, sizeof(float) * (size_t)BS * NSLOT * MDIM,
                 hipMemcpyDeviceToDevice, stream);
  hipMemcpyAsync(out_rw, rw_in, sizeof(float) * (size_t)BS * RHEAD * NSLOT,
                 hipMemcpyDeviceToDevice, stream);
  hipMemcpyAsync(out_uw, uw_in, sizeof(float) * (size_t)BS * NSLOT,
                 hipMemcpyDeviceToDevice, stream);
  hipMemsetAsync(hidden, 0, sizeof(float) * (size_t)BS * HDIM, stream);
  hipMemsetAsync(cell,   0, sizeof(float) * (size_t)BS * HDIM, stream);

  k_pad_wl<<<(KPAD * G4H) / 256, 256, 0, stream>>>(Wl, Wlp);

  for (int t = 0; t < SEQ; ++t) {
    k_build_a<<<BS, 256, 0, stream>>>(x, y, hidden, Abuf, t);
    k_lstm_gates<<<dim3(BS / 16, G4H / 16), 32, 0, stream>>>(Abuf, Wlp, bl, gates);
    k_lstm_point<<<BS, HDIM, 0, stream>>>(gates, hidden, cell, outputs, t);
    k_prep<<<1, BS, 0, stream>>>(out_uw, idx, norms);
    k_query<<<dim3(BS / 16, HDIM / 16), 32, 0, stream>>>(hidden, Wm, bm, query, norms);
    k_wwfill<<<(BS * RHEAD * NSLOT) / 256, 256, 0, stream>>>(out_rw, alpha, idx, wwb);
    k_memupd<<<BS * NSLOT, MDIM, 0, stream>>>(out_mem, wwb, query, norms);
    k_logits<<<dim3(BS, NSLOT / 16), 32, 0, stream>>>(query, out_mem, norms, logits);
    k_softmax_rw<<<BS * RHEAD, 256, 0, stream>>>(logits, out_rw);
    k_uw<<<(BS * NSLOT) / 256, 256, 0, stream>>>(out_uw, out_rw, wwb);
    k_read<<<BS, MDIM, 0, stream>>>(out_mem, out_rw, outputs, t);
  }
  k_final_gemm<<<dim3(BS * SEQ / 16, OUTD / 16), 32, 0, stream>>>(outputs, Wf, bfv, fin);
  k_final_softmax<<<BS * SEQ, OUTD, 0, stream>>>(fin, out_probs);
}